// RGCN_V_Encoder_61881888801359
// MI455X (gfx1250) — compile-verified
//
#include <hip/hip_runtime.h>
#include <hip/hip_bf16.h>

#define Nn 50000
#define Ee 800000
#define Rr 8
#define NB 30
#define IND 128
#define HIDD 128
#define ZD 64

typedef __attribute__((ext_vector_type(2))) float v2f;
typedef __attribute__((ext_vector_type(8))) float v8f;

// ---------------- edge count per (dst, relation) ----------------
__global__ void count_edges_k(const int* __restrict__ dst, const int* __restrict__ et,
                              unsigned* __restrict__ cnt, int E_) {
    int e = blockIdx.x * blockDim.x + threadIdx.x;
    if (e < E_) atomicAdd(&cnt[(size_t)dst[e] * Rr + et[e]], 1u);
}

// ---------------- W[r] = sum_b comp[r,b]*basis[b]; W[R] = root ----------------
__global__ void build_w_k(const float* __restrict__ comp, const float* __restrict__ basis,
                          const float* __restrict__ root, float* __restrict__ W,
                          int IN, int OUT) {
    int tid = blockIdx.x * blockDim.x + threadIdx.x;
    int per = IN * OUT;
    if (tid >= (Rr + 1) * per) return;
    int r = tid / per;
    int io = tid % per;
    if (r < Rr) {
        float s = 0.f;
        for (int b = 0; b < NB; ++b)
            s += comp[r * NB + b] * basis[(size_t)b * per + io];
        W[tid] = s;
    } else {
        W[tid] = root[io];
    }
}

// ---------------- WMMA fp32 GEMM: C[MxOUT] = A[MxK] * B[KxOUT], M%16==0 ----------------
// One wave per 16-row block. B staged through LDS in <=64-column chunks, swizzled into
// (k, k+1) element pairs so each WMMA B fragment is a single aligned ds_load_b64
// (no repack movs in the inner loop). A fragments preloaded into VGPRs.
template <int K, int OUT>
__global__ __launch_bounds__(256) void gemm_wmma_k(const float* __restrict__ A,
                                                   const float* __restrict__ Bm,
                                                   float* __restrict__ C, int M) {
    constexpr int CH = (OUT > 64) ? 64 : OUT;   // LDS column chunk
    constexpr int CH4 = CH / 4;
    // Bs layout: pair p = k>>1, col c, j = k&1  ->  Bs[(p*CH + c)*2 + j]
    __shared__ float Bs[K * CH];

    const int wave = (blockIdx.x * 256 + threadIdx.x) >> 5;
    const int lane = threadIdx.x & 31;
    const int hf   = lane >> 4;      // half-wave: selects K pair (k+0..1 vs k+2..3)
    const int lr   = lane & 15;
    const int row0 = wave * 16;
    const bool active = (row0 < M);

    // Preload A fragments: a[kk] = {A[row][4kk+2hf], A[row][4kk+2hf+1]}
    v2f afrag[K / 4];
    if (active) {
        const float* a_row = A + (size_t)(row0 + lr) * K + 2 * hf;
#pragma unroll
        for (int kk = 0; kk < K / 4; ++kk)
            afrag[kk] = *(const v2f*)(a_row + kk * 4);
    }

#pragma unroll
    for (int ho = 0; ho < OUT / CH; ++ho) {
        __syncthreads();
        // cooperative 128-bit global loads, swizzled stores into paired LDS layout
        for (int idx = threadIdx.x; idx < K * CH4; idx += 256) {
            int k  = idx / CH4;
            int c4 = idx % CH4;
            float4 f = *(const float4*)(Bm + (size_t)k * OUT + ho * CH + c4 * 4);
            float* bp = Bs + ((size_t)(k >> 1) * CH + c4 * 4) * 2 + (k & 1);
            bp[0] = f.x;
            bp[2] = f.y;
            bp[4] = f.z;
            bp[6] = f.w;
        }
        __syncthreads();
        if (!active) continue;

#pragma unroll
        for (int t = 0; t < CH / 16; ++t) {
            v8f acc = {};
#pragma unroll
            for (int kk = 0; kk < K / 4; ++kk) {
                // rows (4kk+2hf, 4kk+2hf+1) -> pair p = 2kk+hf : one ds_load_b64
                v2f b = *(const v2f*)(Bs + ((size_t)(2 * kk + hf) * CH + t * 16 + lr) * 2);
                acc = __builtin_amdgcn_wmma_f32_16x16x4_f32(
                    false, afrag[kk], false, b, (short)0, acc, false, false);
            }
            // store: VGPR v, half hf -> row row0 + v + 8*hf, col = ho*CH + t*16 + lr
            float* crow = C + (size_t)(row0 + 8 * hf) * OUT + ho * CH + t * 16 + lr;
#pragma unroll
            for (int v = 0; v < 8; ++v)
                crow[(size_t)v * OUT] = acc[v];
        }
    }
}

// ---------------- per-relation edge scatter with mean norm ----------------
__global__ void scatter_rel_k(const float* __restrict__ H, const int* __restrict__ src,
                              const int* __restrict__ dst, const int* __restrict__ et,
                              const unsigned* __restrict__ cnt, float* __restrict__ agg,
                              int E_, int OUT, int r) {
    int out4 = OUT >> 2;
    int tid = blockIdx.x * blockDim.x + threadIdx.x;
    int e = tid / out4;
    int c = tid % out4;
    if (e >= E_) return;
    if (et[e] != r) return;
    int d = dst[e];
    float w = 1.0f / (float)cnt[(size_t)d * Rr + r];
    float4 m = ((const float4*)(H + (size_t)src[e] * OUT))[c];
    float* ap = agg + (size_t)d * OUT + c * 4;
    atomicAdd(ap + 0, m.x * w);
    atomicAdd(ap + 1, m.y * w);
    atomicAdd(ap + 2, m.z * w);
    atomicAdd(ap + 3, m.w * w);
}

// ---------------- out = (agg + rootout + bias)  [optional leaky_relu 0.01] ----------------
__global__ void finish_layer_k(const float* __restrict__ agg, const float* __restrict__ ro,
                               const float* __restrict__ bias, float* __restrict__ out,
                               int M, int OUT, int lrelu) {
    int tid = blockIdx.x * blockDim.x + threadIdx.x;
    if (tid >= M * OUT) return;
    float v = agg[tid] + ro[tid] + bias[tid % OUT];
    if (lrelu) v = (v > 0.f) ? v : 0.01f * v;
    out[tid] = v;
}

// ---------------- z = mu + eps * exp(logstd) ----------------
__global__ void reparam_k(const float* __restrict__ mu, const float* __restrict__ ls,
                          const float* __restrict__ eps, float* __restrict__ z, int n) {
    int i = blockIdx.x * blockDim.x + threadIdx.x;
    if (i < n) z[i] = mu[i] + eps[i] * __expf(ls[i]);
}

extern "C" void kernel_launch(void* const* d_in, const int* in_sizes, int n_in,
                              void* d_out, int out_size, void* d_ws, size_t ws_size,
                              hipStream_t stream) {
    const float* x       = (const float*)d_in[0];
    const int*   ei      = (const int*)d_in[1];
    const int*   et      = (const int*)d_in[2];
    const float* eps     = (const float*)d_in[3];
    const float* basis1  = (const float*)d_in[4];
    const float* comp1   = (const float*)d_in[5];
    const float* root1   = (const float*)d_in[6];
    const float* bias1   = (const float*)d_in[7];
    const float* basisMu = (const float*)d_in[8];
    const float* compMu  = (const float*)d_in[9];
    const float* rootMu  = (const float*)d_in[10];
    const float* biasMu  = (const float*)d_in[11];
    const float* basisLs = (const float*)d_in[12];
    const float* compLs  = (const float*)d_in[13];
    const float* rootLs  = (const float*)d_in[14];
    const float* biasLs  = (const float*)d_in[15];

    const int* src = ei;
    const int* dst = ei + Ee;

    // bump allocator on workspace (~105 MB total)
    char* ws = (char*)d_ws;
    auto alloc = [&](size_t bytes) -> void* {
        void* p = (void*)ws;
        ws += (bytes + 255) & ~(size_t)255;
        return p;
    };
    unsigned* cnt  = (unsigned*)alloc((size_t)Nn * Rr * 4);
    float* W1      = (float*)alloc((size_t)(Rr + 1) * IND * HIDD * 4);
    float* Wmu     = (float*)alloc((size_t)(Rr + 1) * HIDD * ZD * 4);
    float* Wls     = (float*)alloc((size_t)(Rr + 1) * HIDD * ZD * 4);
    float* Hr      = (float*)alloc((size_t)Nn * HIDD * 4);   // per-relation GEMM out (reused)
    float* rootout = (float*)alloc((size_t)Nn * HIDD * 4);
    float* agg     = (float*)alloc((size_t)Nn * HIDD * 4);
    float* h       = (float*)alloc((size_t)Nn * HIDD * 4);

    float* zout  = (float*)d_out;
    float* muout = zout + (size_t)Nn * ZD;
    float* lsout = zout + 2 * (size_t)Nn * ZD;

    const int gemmBlocks = ((Nn / 16) * 32 + 255) / 256;   // one wave per 16 rows

    // counts (shared by all layers)
    hipMemsetAsync(cnt, 0, (size_t)Nn * Rr * 4, stream);
    count_edges_k<<<(Ee + 255) / 256, 256, 0, stream>>>(dst, et, cnt, Ee);

    // materialize relation weights (+ root as slot R)
    build_w_k<<<((Rr + 1) * IND * HIDD + 255) / 256, 256, 0, stream>>>(comp1, basis1, root1, W1, IND, HIDD);
    build_w_k<<<((Rr + 1) * HIDD * ZD + 255) / 256, 256, 0, stream>>>(compMu, basisMu, rootMu, Wmu, HIDD, ZD);
    build_w_k<<<((Rr + 1) * HIDD * ZD + 255) / 256, 256, 0, stream>>>(compLs, basisLs, rootLs, Wls, HIDD, ZD);

    // ---- layer 1: h = lrelu(mean-agg_r(X@W_r) + X@root + bias) ----
    hipMemsetAsync(agg, 0, (size_t)Nn * HIDD * 4, stream);
    for (int r = 0; r < Rr; ++r) {
        gemm_wmma_k<IND, HIDD><<<gemmBlocks, 256, 0, stream>>>(x, W1 + (size_t)r * IND * HIDD, Hr, Nn);
        scatter_rel_k<<<((size_t)Ee * (HIDD / 4) + 255) / 256, 256, 0, stream>>>(
            Hr, src, dst, et, cnt, agg, Ee, HIDD, r);
    }
    gemm_wmma_k<IND, HIDD><<<gemmBlocks, 256, 0, stream>>>(x, W1 + (size_t)Rr * IND * HIDD, rootout, Nn);
    finish_layer_k<<<(Nn * HIDD + 255) / 256, 256, 0, stream>>>(agg, rootout, bias1, h, Nn, HIDD, 1);

    // ---- mu layer ----
    hipMemsetAsync(agg, 0, (size_t)Nn * ZD * 4, stream);
    for (int r = 0; r < Rr; ++r) {
        gemm_wmma_k<HIDD, ZD><<<gemmBlocks, 256, 0, stream>>>(h, Wmu + (size_t)r * HIDD * ZD, Hr, Nn);
        scatter_rel_k<<<((size_t)Ee * (ZD / 4) + 255) / 256, 256, 0, stream>>>(
            Hr, src, dst, et, cnt, agg, Ee, ZD, r);
    }
    gemm_wmma_k<HIDD, ZD><<<gemmBlocks, 256, 0, stream>>>(h, Wmu + (size_t)Rr * HIDD * ZD, rootout, Nn);
    finish_layer_k<<<(Nn * ZD + 255) / 256, 256, 0, stream>>>(agg, rootout, biasMu, muout, Nn, ZD, 0);

    // ---- logstd layer ----
    hipMemsetAsync(agg, 0, (size_t)Nn * ZD * 4, stream);
    for (int r = 0; r < Rr; ++r) {
        gemm_wmma_k<HIDD, ZD><<<gemmBlocks, 256, 0, stream>>>(h, Wls + (size_t)r * HIDD * ZD, Hr, Nn);
        scatter_rel_k<<<((size_t)Ee * (ZD / 4) + 255) / 256, 256, 0, stream>>>(
            Hr, src, dst, et, cnt, agg, Ee, ZD, r);
    }
    gemm_wmma_k<HIDD, ZD><<<gemmBlocks, 256, 0, stream>>>(h, Wls + (size_t)Rr * HIDD * ZD, rootout, Nn);
    finish_layer_k<<<(Nn * ZD + 255) / 256, 256, 0, stream>>>(agg, rootout, biasLs, lsout, Nn, ZD, 0);

    // ---- reparameterization ----
    reparam_k<<<(Nn * ZD + 255) / 256, 256, 0, stream>>>(muout, lsout, eps, zout, Nn * ZD);
}